// HungarianMatcher_26474178413313
// MI455X (gfx1250) — compile-verified
//
#include <hip/hip_runtime.h>
#include <hip/hip_bf16.h>
#include <math.h>
#include <stdint.h>

#define BN   2048   // B*N rows/cols of full cost matrix
#define NN   256    // N
#define NB   8      // B
#define DD   128    // descriptor dim
#define FROW 130    // feature row stride (2 kp + 128 desc)
#define LDB  132    // padded LDS stride for B descriptors (16B aligned, conflict-free)

typedef float v2f __attribute__((ext_vector_type(2)));
typedef float v8f __attribute__((ext_vector_type(8)));
typedef uint32_t u32x4 __attribute__((ext_vector_type(4)));
typedef uint32_t u32x8 __attribute__((ext_vector_type(8)));

// ---------------------------------------------------------------------------
// Kernel 1: keypoint transform + int truncation -> pts1, pts2 (ws)
// ---------------------------------------------------------------------------
__global__ __launch_bounds__(256) void hm_prep(
    const float* __restrict__ theta,
    const float* __restrict__ feat1, const float* __restrict__ feat2,
    const int* __restrict__ wptr, const int* __restrict__ hptr,
    float* __restrict__ pts1, float* __restrict__ pts2) {
  int i = blockIdx.x * blockDim.x + threadIdx.x;
  if (i >= BN) return;
  float th = theta[0];
  float c = cosf(th), s = sinf(th);
  float W = (float)wptr[0], H = (float)hptr[0];
  const float* f1 = feat1 + (size_t)i * FROW;
  const float* f2 = feat2 + (size_t)i * FROW;
  // pts1 = int(my_tf_kp)
  pts1[i * 2 + 0] = (float)(int)f2[0];
  pts1[i * 2 + 1] = (float)(int)f2[1];
  // transformed my_kp -> o -> pts2
  float xk = f1[0] - 0.5f * W;
  float yk = -f1[1] + 0.5f * H;
  float cx = rintf(c * xk - s * yk);   // round-half-even, matches jnp.round
  float cy = rintf(s * xk + c * yk);
  float ox = cx + 0.5f * W;
  float oy = -cy + 0.5f * H;
  pts2[i * 2 + 0] = (float)(int)ox;
  pts2[i * 2 + 1] = (float)(int)oy;
}

// ---------------------------------------------------------------------------
// Kernel 2: 8-point fundamental matrix + q = p1h @ F precompute
// ---------------------------------------------------------------------------
__device__ double hm_breduce(double val, double* buf) {
  int tid = threadIdx.x;
  buf[tid] = val;
  __syncthreads();
  for (int s = 128; s > 0; s >>= 1) {
    if (tid < s) buf[tid] += buf[tid + s];
    __syncthreads();
  }
  double r = buf[0];
  __syncthreads();
  return r;
}

// Cyclic Jacobi eigensolver for symmetric n x n (n <= 9), columns of V = eigvecs
__device__ void hm_jacobi(double* M, double* V, int n) {
  for (int i = 0; i < n; i++)
    for (int j = 0; j < n; j++) V[i * n + j] = (i == j) ? 1.0 : 0.0;
  for (int sweep = 0; sweep < 40; ++sweep) {
    double off = 0.0;
    for (int p = 0; p < n - 1; p++)
      for (int q = p + 1; q < n; q++) off += M[p * n + q] * M[p * n + q];
    if (off < 1e-26) break;
    for (int p = 0; p < n - 1; p++) {
      for (int q = p + 1; q < n; q++) {
        double apq = M[p * n + q];
        if (fabs(apq) < 1e-300) continue;
        double app = M[p * n + p], aqq = M[q * n + q];
        double tau = (aqq - app) / (2.0 * apq);
        double t = (tau >= 0.0 ? 1.0 : -1.0) / (fabs(tau) + sqrt(1.0 + tau * tau));
        double cth = 1.0 / sqrt(1.0 + t * t), sth = t * cth;
        for (int k = 0; k < n; k++) {
          double mkp = M[k * n + p], mkq = M[k * n + q];
          M[k * n + p] = cth * mkp - sth * mkq;
          M[k * n + q] = sth * mkp + cth * mkq;
        }
        for (int k = 0; k < n; k++) {
          double mpk = M[p * n + k], mqk = M[q * n + k];
          M[p * n + k] = cth * mpk - sth * mqk;
          M[q * n + k] = sth * mpk + cth * mqk;
        }
        for (int k = 0; k < n; k++) {
          double vkp = V[k * n + p], vkq = V[k * n + q];
          V[k * n + p] = cth * vkp - sth * vkq;
          V[k * n + q] = sth * vkp + cth * vkq;
        }
      }
    }
  }
}

__global__ __launch_bounds__(256) void hm_solveF(
    const float* __restrict__ pts1, const float* __restrict__ pts2,
    float* __restrict__ qout) {
  __shared__ double buf[256];
  __shared__ double sATA[45];
  __shared__ float sF[9];
  int tid = threadIdx.x;
  // phase 1: means
  double s1x = 0, s1y = 0, s2x = 0, s2y = 0;
  for (int i = tid; i < BN; i += 256) {
    s1x += pts1[i * 2]; s1y += pts1[i * 2 + 1];
    s2x += pts2[i * 2]; s2y += pts2[i * 2 + 1];
  }
  double m1x = hm_breduce(s1x, buf) / BN;
  double m1y = hm_breduce(s1y, buf) / BN;
  double m2x = hm_breduce(s2x, buf) / BN;
  double m2y = hm_breduce(s2y, buf) / BN;
  // phase 2: scales s = sqrt(2)/mean(dist)
  double d1 = 0, d2 = 0;
  for (int i = tid; i < BN; i += 256) {
    double dx = pts1[i * 2] - m1x, dy = pts1[i * 2 + 1] - m1y;
    d1 += sqrt(dx * dx + dy * dy);
    dx = pts2[i * 2] - m2x; dy = pts2[i * 2 + 1] - m2y;
    d2 += sqrt(dx * dx + dy * dy);
  }
  double sc1 = 1.4142135623730951 / (hm_breduce(d1, buf) / BN);
  double sc2 = 1.4142135623730951 / (hm_breduce(d2, buf) / BN);
  // phase 3: accumulate A^T A (upper triangle, 45 entries)
  if (tid < 45) sATA[tid] = 0.0;
  __syncthreads();
  for (int i = tid; i < BN; i += 256) {
    double u1 = (pts1[i * 2] - m1x) * sc1, v1 = (pts1[i * 2 + 1] - m1y) * sc1;
    double u2 = (pts2[i * 2] - m2x) * sc2, v2 = (pts2[i * 2 + 1] - m2y) * sc2;
    double a[9] = {u2 * u1, u2 * v1, u2, v2 * u1, v2 * v1, v2, u1, v1, 1.0};
    int idx = 0;
    for (int p = 0; p < 9; p++)
      for (int q = p; q < 9; q++)
        atomicAdd(&sATA[idx++], a[p] * a[q]);
  }
  __syncthreads();
  if (tid == 0) {
    double M[81], V[81];
    int idx = 0;
    for (int p = 0; p < 9; p++)
      for (int q = p; q < 9; q++) { double v = sATA[idx++]; M[p * 9 + q] = v; M[q * 9 + p] = v; }
    hm_jacobi(M, V, 9);
    int k = 0; double best = M[0];
    for (int i2 = 1; i2 < 9; i2++) if (M[i2 * 9 + i2] < best) { best = M[i2 * 9 + i2]; k = i2; }
    double Fm[9];
    for (int e = 0; e < 9; e++) Fm[e] = V[e * 9 + k];   // smallest right-singular vec of A
    // rank-2 projection: U diag(s1,s2,0) Vt == Fm (I - v3 v3^T), v3 = min eigvec of Fm^T Fm
    double G[9], VG[9];
    for (int a2 = 0; a2 < 3; a2++)
      for (int b2 = 0; b2 < 3; b2++) {
        double acc = 0;
        for (int r = 0; r < 3; r++) acc += Fm[r * 3 + a2] * Fm[r * 3 + b2];
        G[a2 * 3 + b2] = acc;
      }
    hm_jacobi(G, VG, 3);
    int k3 = 0; best = G[0];
    for (int i2 = 1; i2 < 3; i2++) if (G[i2 * 3 + i2] < best) { best = G[i2 * 3 + i2]; k3 = i2; }
    double v3[3] = {VG[0 * 3 + k3], VG[1 * 3 + k3], VG[2 * 3 + k3]};
    double Fv[3];
    for (int r = 0; r < 3; r++)
      Fv[r] = Fm[r * 3 + 0] * v3[0] + Fm[r * 3 + 1] * v3[1] + Fm[r * 3 + 2] * v3[2];
    double F2[9];
    for (int r = 0; r < 3; r++)
      for (int c2 = 0; c2 < 3; c2++) F2[r * 3 + c2] = Fm[r * 3 + c2] - Fv[r] * v3[c2];
    // denormalize: T2^T F2 T1, then / F[2][2]
    double T1[9] = {sc1, 0, -sc1 * m1x, 0, sc1, -sc1 * m1y, 0, 0, 1};
    double T2[9] = {sc2, 0, -sc2 * m2x, 0, sc2, -sc2 * m2y, 0, 0, 1};
    double t[9];
    for (int r = 0; r < 3; r++)
      for (int c2 = 0; c2 < 3; c2++) {
        double acc = 0;
        for (int k2 = 0; k2 < 3; k2++) acc += F2[r * 3 + k2] * T1[k2 * 3 + c2];
        t[r * 3 + c2] = acc;
      }
    double F3[9];
    for (int r = 0; r < 3; r++)
      for (int c2 = 0; c2 < 3; c2++) {
        double acc = 0;
        for (int k2 = 0; k2 < 3; k2++) acc += T2[k2 * 3 + r] * t[k2 * 3 + c2];
        F3[r * 3 + c2] = acc;
      }
    double inv = 1.0 / F3[8];
    for (int e = 0; e < 9; e++) sF[e] = (float)(F3[e] * inv);
  }
  __syncthreads();
  // phase 5: q_i = p1h_i^T F  (padded to 4 for WMMA K=4)
  float F00 = sF[0], F01 = sF[1], F02 = sF[2];
  float F10 = sF[3], F11 = sF[4], F12 = sF[5];
  float F20 = sF[6], F21 = sF[7], F22 = sF[8];
  for (int i = tid; i < BN; i += 256) {
    float x1 = pts1[i * 2], y1 = pts1[i * 2 + 1];
    qout[i * 4 + 0] = x1 * F00 + y1 * F10 + F20;
    qout[i * 4 + 1] = x1 * F01 + y1 * F11 + F21;
    qout[i * 4 + 2] = x1 * F02 + y1 * F12 + F22;
    qout[i * 4 + 3] = 0.0f;
  }
}

// ---------------------------------------------------------------------------
// TDM helper: async 2D tile load (rows x 128 f32, global row stride 130 f32)
// into LDS, optional post-row padding (D# per CDNA5 ISA).
// Issued once per executing wave; tracked with TENSORcnt.
// ---------------------------------------------------------------------------
__device__ __forceinline__ void tdm_load_2d(uint32_t lds_off, const void* gptr,
                                            uint32_t rows, uint32_t g1s0) {
  uint64_t ga = (uint64_t)(uintptr_t)gptr;
  u32x4 g0;
  g0.x = 1u;                                              // count=1 valid, user mode
  g0.y = lds_off;                                         // lds_addr (bytes)
  g0.z = (uint32_t)ga;                                    // global_addr[31:0]
  g0.w = ((uint32_t)(ga >> 32) & 0x01FFFFFFu) | (2u << 30); // addr[56:32], type=2
  u32x8 g1;
  g1.s0 = g1s0;                       // data_size=4B (+ pad config for B tile)
  g1.s1 = (uint32_t)DD << 16;         // tensor_dim0 = 128 (abar_addr=0)
  g1.s2 = (rows & 0xFFFFu) << 16;     // tensor_dim0[31:16]=0, tensor_dim1 = rows
  g1.s3 = (uint32_t)DD << 16;         // tensor_dim1[31:16]=0, tile_dim0 = 128
  g1.s4 = rows & 0xFFFFu;             // tile_dim1 = rows, tile_dim2 = 0
  g1.s5 = (uint32_t)FROW;             // tensor_dim0_stride = 130 (low 32)
  g1.s6 = 0u;                         // stride0 hi16 = 0, stride1 lo16 = 0
  g1.s7 = 0u;                         // stride1 hi32 = 0
  asm volatile("tensor_load_to_lds %0, %1" :: "s"(g0), "s"(g1) : "memory");
}

// ---------------------------------------------------------------------------
// Kernel 3: C[i][j] = |q_i . p2h_j| (WMMA f32 16x16x4) + L1(desc1_i, desc2_j)
// Block: 256 threads = 8 waves; each wave owns one 16x16 tile; block = 32x64.
// Descriptor tiles staged by the Tensor Data Mover (wave 0 issues, TENSORcnt
// wait, then workgroup barrier); B tile padded by TDM to a 132-float stride
// (pad_interval=6 -> every 128 DWORDs, pad_amount=3 -> 4 DWORDs) so b128 LDS
// reads are bank-conflict free. L1 accumulators match the WMMA f32 C/D layout
// (M = r + 8*(lane>=16), N = lane%16) so the WMMA result merges in-register.
// ---------------------------------------------------------------------------
__global__ __launch_bounds__(256) void hm_cost(
    const float* __restrict__ feat1, const float* __restrict__ feat2,
    const float* __restrict__ pts2, const float* __restrict__ q,
    float* __restrict__ C) {
  __shared__ float ldsA[32 * DD];    // desc1 rows (broadcast reads)
  __shared__ float ldsB[64 * LDB];   // desc2 rows, TDM-padded stride
  int tid = threadIdx.x;
  int rowBase = blockIdx.y * 32;
  int colBase = blockIdx.x * 64;

  if (tid < 32) {   // wave 0 issues both TDM transfers (TDM issues per-wave)
    uint32_t offA = (uint32_t)(uintptr_t)(&ldsA[0]);
    uint32_t offB = (uint32_t)(uintptr_t)(&ldsB[0]);
    tdm_load_2d(offA, feat1 + (size_t)rowBase * FROW + 2, 32u,
                (2u << 16));                                  // 4B elements
    tdm_load_2d(offB, feat2 + (size_t)colBase * FROW + 2, 64u,
                (2u << 16) | (1u << 20) | (6u << 22) | (3u << 25)); // + padding
    __builtin_amdgcn_s_wait_tensorcnt(0);
  }
  __syncthreads();

  int wave = tid >> 5, lane = tid & 31;
  int wr = wave >> 2, wc = wave & 3;     // 2x4 arrangement of 16x16 tiles
  int tr = wr * 16, tc = wc * 16;
  int laneN = lane & 15;
  int hi = lane >> 4;                    // 0 or 1
  int mOff = tr + hi * 8;                // accumulator base row within block

  float acc[8] = {0.f, 0.f, 0.f, 0.f, 0.f, 0.f, 0.f, 0.f};
  const float* bcol = &ldsB[(tc + laneN) * LDB];
  for (int k = 0; k < DD; k += 4) {
    float4 b4 = *reinterpret_cast<const float4*>(&bcol[k]);
#pragma unroll
    for (int r2 = 0; r2 < 8; ++r2) {
      float4 a4 = *reinterpret_cast<const float4*>(&ldsA[(mOff + r2) * DD + k]);
      acc[r2] += fabsf(a4.x - b4.x) + fabsf(a4.y - b4.y) +
                 fabsf(a4.z - b4.z) + fabsf(a4.w - b4.w);
    }
  }

  // WMMA rank-3 kp term: A[m][k] = q[row m][k] (K padded to 4),
  // B[k][n] = p2h[col n][k] = {x2, y2, 1, 0}
  int gm = rowBase + tr + laneN;         // A-matrix row for this lane (M = lane%16)
  int kA = hi * 2;                       // lanes 0-15 hold K=0,1; lanes 16-31 hold K=2,3
  v2f av;
  av.x = q[gm * 4 + kA];
  av.y = q[gm * 4 + kA + 1];
  int gj = colBase + tc + laneN;         // B-matrix col for this lane (N = lane%16)
  v2f bv;
  bv.x = hi ? 1.0f : pts2[gj * 2 + 0];   // rows K=0,1 vs K=2,3 of p2h
  bv.y = hi ? 0.0f : pts2[gj * 2 + 1];
  v8f c0 = {};
  v8f w = __builtin_amdgcn_wmma_f32_16x16x4_f32(
      /*neg_a=*/false, av, /*neg_b=*/false, bv,
      /*c_mod=*/(short)0, c0, /*reuse_a=*/false, /*reuse_b=*/false);

#pragma unroll
  for (int r2 = 0; r2 < 8; ++r2) {
    int m = rowBase + mOff + r2;         // == rowBase + tr + 8*hi + r2 (matches D layout)
    C[(size_t)m * BN + gj] = fabsf(w[r2]) + acc[r2];
  }
}

// ---------------------------------------------------------------------------
// Kernel 4: JV (shortest augmenting path) Hungarian, one block per batch,
// thread-per-column relaxation, LDS argmin tree, float64 duals (matches scipy)
// ---------------------------------------------------------------------------
__global__ __launch_bounds__(256) void hm_hungarian(
    const float* __restrict__ C, int* __restrict__ matches) {
  const int n = NN;
  int b = blockIdx.x, tid = threadIdx.x;
  const float* cost = C + (size_t)(b * n) * BN + (size_t)(b * n);
  __shared__ double u[256], v[256], shortest[256], rv[256];
  __shared__ int path[256], row4col[256], col4row[256], ri[256];
  __shared__ unsigned char remaining[256], inSR[256];
  __shared__ int s_i, s_sink;
  __shared__ double s_minval;

  u[tid] = 0.0; v[tid] = 0.0;
  row4col[tid] = -1; col4row[tid] = -1;
  __syncthreads();

  for (int cur = 0; cur < n; ++cur) {
    shortest[tid] = 1e300;
    remaining[tid] = 1;
    inSR[tid] = 0;
    if (tid == 0) { s_i = cur; s_minval = 0.0; s_sink = -1; }
    __syncthreads();
    while (s_sink == -1) {
      if (tid == 0) inSR[s_i] = 1;
      __syncthreads();
      int i = s_i;
      double mv = s_minval;
      if (remaining[tid]) {
        double r = mv + (double)cost[(size_t)i * BN + tid] - u[i] - v[tid];
        if (r < shortest[tid]) { shortest[tid] = r; path[tid] = i; }
      }
      __syncthreads();
      rv[tid] = remaining[tid] ? shortest[tid] : 1e301;
      ri[tid] = tid;
      __syncthreads();
      for (int s = 128; s > 0; s >>= 1) {
        if (tid < s) {
          if (rv[tid + s] < rv[tid] ||
              (rv[tid + s] == rv[tid] && ri[tid + s] < ri[tid])) {
            rv[tid] = rv[tid + s];
            ri[tid] = ri[tid + s];
          }
        }
        __syncthreads();
      }
      if (tid == 0) {
        int jp = ri[0];
        s_minval = rv[0];
        remaining[jp] = 0;
        if (row4col[jp] == -1) s_sink = jp;
        else s_i = row4col[jp];
      }
      __syncthreads();
    }
    // dual updates (before augmentation, per scipy)
    double mv = s_minval;
    if (tid == cur) u[tid] += mv;
    else if (inSR[tid]) u[tid] += mv - shortest[col4row[tid]];
    if (!remaining[tid]) v[tid] -= mv - shortest[tid];
    __syncthreads();
    // augment along alternating path (serial)
    if (tid == 0) {
      int j = s_sink;
      for (;;) {
        int i2 = path[j];
        row4col[j] = i2;
        int tmp = col4row[i2];
        col4row[i2] = j;
        j = tmp;
        if (i2 == cur) break;
      }
    }
    __syncthreads();
  }
  // matches[b][0][n] = n ; matches[b][1][n] = col4row[n]
  matches[b * 2 * n + tid] = tid;
  matches[b * 2 * n + n + tid] = col4row[tid];
}

// ---------------------------------------------------------------------------
extern "C" void kernel_launch(void* const* d_in, const int* in_sizes, int n_in,
                              void* d_out, int out_size, void* d_ws, size_t ws_size,
                              hipStream_t stream) {
  const float* theta = (const float*)d_in[0];
  const float* feat1 = (const float*)d_in[1];   // my_feature   (B,N,130)
  const float* feat2 = (const float*)d_in[2];   // my_tf_feature(B,N,130)
  const int* wptr = (const int*)d_in[3];
  const int* hptr = (const int*)d_in[4];

  float* ws = (float*)d_ws;
  float* pts1 = ws;                 // 2048*2
  float* pts2 = ws + 2 * BN;        // 2048*2
  float* q    = ws + 4 * BN;        // 2048*4 (K padded to 4)

  float* Cout = (float*)d_out;                  // 2048*2048 floats
  int* matches = (int*)(Cout + (size_t)BN * BN);// B*2*N int32

  hm_prep<<<BN / 256, 256, 0, stream>>>(theta, feat1, feat2, wptr, hptr, pts1, pts2);
  hm_solveF<<<1, 256, 0, stream>>>(pts1, pts2, q);
  hm_cost<<<dim3(BN / 64, BN / 32), 256, 0, stream>>>(feat1, feat2, pts2, q, Cout);
  hm_hungarian<<<NB, 256, 0, stream>>>(Cout, matches);
}